// Track_1812476199629
// MI455X (gfx1250) — compile-verified
//
#include <hip/hip_runtime.h>
#include <hip/hip_bf16.h>

#define C_LIGHT 0.29979245f
#define CHUNK   256            // steps buffered per TDM store (6*CHUNK*4 = 6KB tile)

typedef __attribute__((ext_vector_type(4))) unsigned int v4u;
typedef __attribute__((ext_vector_type(8))) int          v8i;
typedef __attribute__((ext_vector_type(4))) int          v4i;

// dr/dt and dp/dt for B = (0, B0 cos(ku z), 0); py' == 0 identically.
__device__ __forceinline__ void undulator_deriv(
    float px, float py, float pz, float z,
    float B0, float ku, float invC2,
    float& rkx, float& rky, float& rkz,
    float& pkx, float& pkz)
{
    float By   = B0 * cosf(ku * z);
    float n2   = fmaf(fmaf(px, px, fmaf(py, py, pz * pz)), invC2, 1.0f);
    float invg = rsqrtf(n2);                 // 1/gamma
    rkx = px * invg;
    rky = py * invg;
    rkz = pz * invg;
    float s = By * invg;
    pkx =  pz * s;                           // -(p x B).x / g
    pkz = -px * s;                           // -(p x B).z / g
}

__global__ __launch_bounds__(32) void undulator_track(
    const float* __restrict__ time,
    const float* __restrict__ r0v,
    const float* __restrict__ d0v,
    const float* __restrict__ gamma_p,
    const float* __restrict__ B0_p,
    const float* __restrict__ ku_p,
    float* __restrict__ out, int N)
{
    __shared__ float buf[2][6 * CHUNK];      // double-buffered [6][CHUNK] tiles

    const float dt   = time[1] - time[0];
    const float dt2  = 0.5f * dt;
    const float dt6  = dt * (1.0f / 6.0f);
    const float B0   = B0_p[0];
    const float ku   = ku_p[0];
    const float gam  = gamma_p[0];
    const float C2   = C_LIGHT * C_LIGHT;
    const float invC2 = 1.0f / C2;

    float rx = r0v[0], ry = r0v[1], rz = r0v[2];
    float dx = d0v[0], dy = d0v[1], dz = d0v[2];
    float pmag = C_LIGHT * sqrtf(gam * gam - 1.0f);
    float dinv = rsqrtf(fmaf(dx, dx, fmaf(dy, dy, dz * dz)));
    float px = pmag * dinv * dx;
    float py = pmag * dinv * dy;             // invariant under this field
    float pz = pmag * dinv * dz;

    const int lane   = (int)threadIdx.x;
    const int nchunk = N / CHUNK;
    const unsigned long long outAddr = (unsigned long long)(size_t)out;
    const unsigned int ldsBase = (unsigned int)(size_t)(&buf[0][0]);
    const unsigned int strideN = (unsigned int)N;

    for (int c = 0; c < nchunk; ++c) {
        const int b = c & 1;
        // buffer b was handed to the TDM two chunks ago; <=1 outstanding => it's free
        __builtin_amdgcn_s_wait_tensorcnt(1);

        float* lbuf = &buf[b][0];
        #pragma unroll 1
        for (int i = 0; i < CHUNK; ++i) {
            // ---- record state at this timestep (r, beta = p/sqrt(c^2+|p|^2)) ----
            float pn2  = fmaf(px, px, fmaf(py, py, pz * pz));
            float invb = rsqrtf(C2 + pn2);
            if (lane == 0) {
                lbuf[0 * CHUNK + i] = rx;
                lbuf[1 * CHUNK + i] = ry;
                lbuf[2 * CHUNK + i] = rz;
                lbuf[3 * CHUNK + i] = px * invb;
                lbuf[4 * CHUNK + i] = py * invb;
                lbuf[5 * CHUNK + i] = pz * invb;
            }
            // ---- RK4 step ----
            float r1x, r1y, r1z, p1x, p1z;
            undulator_deriv(px, py, pz, rz, B0, ku, invC2, r1x, r1y, r1z, p1x, p1z);
            float r2x, r2y, r2z, p2x, p2z;
            undulator_deriv(fmaf(p1x, dt2, px), py, fmaf(p1z, dt2, pz),
                            fmaf(r1z, dt2, rz), B0, ku, invC2, r2x, r2y, r2z, p2x, p2z);
            float r3x, r3y, r3z, p3x, p3z;
            undulator_deriv(fmaf(p2x, dt2, px), py, fmaf(p2z, dt2, pz),
                            fmaf(r2z, dt2, rz), B0, ku, invC2, r3x, r3y, r3z, p3x, p3z);
            float r4x, r4y, r4z, p4x, p4z;
            undulator_deriv(fmaf(p3x, dt, px), py, fmaf(p3z, dt, pz),
                            fmaf(r3z, dt, rz), B0, ku, invC2, r4x, r4y, r4z, p4x, p4z);

            rx = fmaf(dt6, r1x + 2.0f * (r2x + r3x) + r4x, rx);
            ry = fmaf(dt6, r1y + 2.0f * (r2y + r3y) + r4y, ry);
            rz = fmaf(dt6, r1z + 2.0f * (r2z + r3z) + r4z, rz);
            px = fmaf(dt6, p1x + 2.0f * (p2x + p3x) + p4x, px);
            pz = fmaf(dt6, p1z + 2.0f * (p2z + p3z) + p4z, pz);
            // py unchanged (dp/dt.y == 0 for this field)
        }

        // make lane-0 DS writes visible before the TDM reads the tile
        asm volatile("s_wait_dscnt 0" ::: "memory");

        // ---- Tensor DMA: LDS [6][CHUNK] tile -> global [6][N] at column c*CHUNK ----
        unsigned long long ga = outAddr + (unsigned long long)c * (CHUNK * 4ull);
        unsigned int ga_lo = (unsigned int)(ga & 0xFFFFFFFFull);
        unsigned int ga_hi = (unsigned int)((ga >> 32) & 0x1FFFFFFull);
        unsigned int ldsA  = ldsBase + (unsigned int)(b * 6 * CHUNK * 4);

        v4u g0 = { 1u,                          // count=1 (valid), is_restore=0, gather off
                   ldsA,                        // lds_addr (bytes)
                   ga_lo,                       // global_addr[31:0]
                   ga_hi | (2u << 30) };        // global_addr[56:32] | type=2 ("image")

        v8i g1 = { (int)(2u << 16),                                   // data_size=4B
                   (int)((strideN & 0xFFFFu) << 16),                  // tensor_dim0[15:0]
                   (int)(((strideN >> 16) & 0xFFFFu) | (6u << 16)),   // dim0 hi | tensor_dim1=6
                   (int)((unsigned)CHUNK << 16),                      // dim1 hi | tile_dim0=CHUNK
                   6,                                                 // tile_dim1=6, tile_dim2=0
                   (int)strideN,                                      // tensor_dim0_stride lo (=N elems)
                   0,                                                 // dim0_stride hi | dim1_stride lo
                   0 };
        v4i g2 = { 0, 0, 0, 0 };                // 2D tensor: dims 2/3 unused
        v4i g3 = { 0, 0, 0, 0 };

#if __clang_major__ >= 23
        v8i g4 = { 0, 0, 0, 0, 0, 0, 0, 0 };
        __builtin_amdgcn_tensor_store_from_lds(g0, g1, g2, g3, g4, 0);
#else
        __builtin_amdgcn_tensor_store_from_lds(g0, g1, g2, g3, 0);
#endif
    }
    // S_ENDPGM performs an implicit wait-idle (TENSORcnt drains before exit)
}

extern "C" void kernel_launch(void* const* d_in, const int* in_sizes, int n_in,
                              void* d_out, int out_size, void* d_ws, size_t ws_size,
                              hipStream_t stream) {
    (void)n_in; (void)out_size; (void)d_ws; (void)ws_size;
    const float* time = (const float*)d_in[0];
    const float* r0   = (const float*)d_in[1];
    const float* d0   = (const float*)d_in[2];
    const float* gam  = (const float*)d_in[3];
    const float* B0   = (const float*)d_in[4];
    const float* ku   = (const float*)d_in[5];
    int N = in_sizes[0];                       // 1048576, divisible by CHUNK
    undulator_track<<<dim3(1), dim3(32), 0, stream>>>(
        time, r0, d0, gam, B0, ku, (float*)d_out, N);
}